// AttentionGAT_30709016166928
// MI455X (gfx1250) — compile-verified
//
#include <hip/hip_runtime.h>
#include <hip/hip_bf16.h>

typedef float v2f __attribute__((ext_vector_type(2)));
typedef float v8f __attribute__((ext_vector_type(8)));

#define HH 8
#define DD 64
#define ALPHA 0.2f
#define EPSV 1e-9f

// sign-correct float atomic max via integer atomics (works for mixed signs,
// buffers initialized to -inf)
__device__ __forceinline__ void atomicMaxF(float* addr, float v) {
  if (v >= 0.0f) {
    atomicMax((int*)addr, __float_as_int(v));
  } else {
    atomicMin((unsigned int*)addr, (unsigned int)__float_as_int(v));
  }
}

// -----------------------------------------------------------------------------
// Pass 1: a_self[r] = X[r,:] . k_self[:, r%8], a_adjc[r] = X[r,:] . k_adjc[:, r%8]
// X flattened as [R = N*H, 64] fp32 row-major. One wave per 16-row tile.
// B (64x16): cols 0..7 = k_self heads, cols 8..15 = k_adjc heads.
// Accumulate D = A(16x64) x B(64x16) with 16 chained V_WMMA_F32_16X16X4_F32.
// D[i, i%8] = a_self(row i), D[i, 8+i%8] = a_adjc(row i).
// Per the 16x16 f32 C/D layout, lane L owns exactly one useful value: acc[L&7].
// -----------------------------------------------------------------------------
__global__ void gat_logits_wmma(const float* __restrict__ X,
                                const float* __restrict__ kself,
                                const float* __restrict__ kadjc,
                                float* __restrict__ a_self,
                                float* __restrict__ a_adjc,
                                int R) {
  const int lane       = threadIdx.x & 31;
  const int waveInBlk  = threadIdx.x >> 5;
  const int wavesPerBl = blockDim.x >> 5;
  const int waveId     = blockIdx.x * wavesPerBl + waveInBlk;
  const int numWaves   = gridDim.x * wavesPerBl;

  const int col  = lane & 15;          // B column this lane holds
  const int koff = (lane >> 4) << 1;   // K sub-offset: 0 (lanes 0-15), 2 (lanes 16-31)
  const int hh   = col & 7;
  const float* kptr = (col < 8) ? kself : kadjc;

  // Loop-invariant B fragments: step k0 covers K = k0*4 .. k0*4+3
  v2f Bf[16];
#pragma unroll
  for (int k0 = 0; k0 < 16; ++k0) {
    const int kk = (k0 << 2) + koff;
    v2f b;
    b.x = kptr[kk * HH + hh];
    b.y = kptr[(kk + 1) * HH + hh];
    Bf[k0] = b;
  }

  const int tiles      = R >> 4;
  const int j          = lane & 7;
  const bool isSelf    = ((lane & 15) < 8);
  const int rowInTile  = ((lane >> 4) << 3) + j;
  float* const dstBase = isSelf ? a_self : a_adjc;

  for (int t = waveId; t < tiles; t += numWaves) {
    const int r0 = t << 4;
    const float* xrow = X + (size_t)(r0 + (lane & 15)) * DD + koff;

    const int tn = t + numWaves;          // prefetch next tile's row
    if (tn < tiles) {
      __builtin_prefetch(X + (size_t)((tn << 4) + (lane & 15)) * DD + koff, 0, 3);
    }

    v8f acc = {0.f, 0.f, 0.f, 0.f, 0.f, 0.f, 0.f, 0.f};
#pragma unroll
    for (int k0 = 0; k0 < 16; ++k0) {
      v2f a = *(const v2f*)(xrow + (k0 << 2));
      acc = __builtin_amdgcn_wmma_f32_16x16x4_f32(
          /*neg_a=*/false, a, /*neg_b=*/false, Bf[k0],
          /*c_mod=*/(short)0, acc, /*reuse_a=*/false, /*reuse_b=*/false);
    }

    float val = acc[0];
    if (j == 1) val = acc[1];
    if (j == 2) val = acc[2];
    if (j == 3) val = acc[3];
    if (j == 4) val = acc[4];
    if (j == 5) val = acc[5];
    if (j == 6) val = acc[6];
    if (j == 7) val = acc[7];
    dstBase[r0 + rowInTile] = val;
  }

  // scalar remainder rows (R % 16), if any
  const int rem = R & 15;
  if (rem && blockIdx.x == 0 && (int)threadIdx.x < rem) {
    const int r = (tiles << 4) + (int)threadIdx.x;
    const int h = r & (HH - 1);
    float s0 = 0.f, s1 = 0.f;
    for (int d = 0; d < DD; ++d) {
      const float x = X[(size_t)r * DD + d];
      s0 += x * kself[d * HH + h];
      s1 += x * kadjc[d * HH + h];
    }
    a_self[r] = s0;
    a_adjc[r] = s1;
  }
}

__global__ void init_neg_inf(float* __restrict__ p, int n) {
  const int i = blockIdx.x * blockDim.x + threadIdx.x;
  if (i < n) p[i] = -__builtin_inff();
}

__device__ __forceinline__ float edge_logit(const float* __restrict__ a_self,
                                            const float* __restrict__ a_adjc,
                                            int t, int s, int h) {
  float v = a_self[t * HH + h] + a_adjc[s * HH + h];
  return fmaxf(v, ALPHA * v);  // leaky_relu(v, 0.2)
}

// segment_max of leaky-relu logits over targets
__global__ void edge_max1(const int* __restrict__ tg, const int* __restrict__ sc,
                          const float* __restrict__ a_self,
                          const float* __restrict__ a_adjc,
                          float* __restrict__ m, long long total) {
  const long long i = (long long)blockIdx.x * blockDim.x + threadIdx.x;
  if (i >= total) return;
  const int e = (int)(i >> 3), h = (int)(i & 7);
  const int t = tg[e], s = sc[e];
  const float v = edge_logit(a_self, a_adjc, t, s, h);
  atomicMaxF(&m[t * HH + h], v);
}

// segment_max of exp(v - m[target])
__global__ void edge_max2(const int* __restrict__ tg, const int* __restrict__ sc,
                          const float* __restrict__ a_self,
                          const float* __restrict__ a_adjc,
                          const float* __restrict__ m, float* __restrict__ m2,
                          long long total) {
  const long long i = (long long)blockIdx.x * blockDim.x + threadIdx.x;
  if (i >= total) return;
  const int e = (int)(i >> 3), h = (int)(i & 7);
  const int t = tg[e], s = sc[e];
  const float v  = edge_logit(a_self, a_adjc, t, s, h);
  const float ex = __expf(v - m[t * HH + h]);
  atomicMaxF(&m2[t * HH + h], ex);
}

// attn = exp(v - m)/(m2 + eps), then deterministic dropout(0.5)
__global__ void edge_out(const int* __restrict__ tg, const int* __restrict__ sc,
                         const float* __restrict__ a_self,
                         const float* __restrict__ a_adjc,
                         const float* __restrict__ m, const float* __restrict__ m2,
                         float* __restrict__ out, long long total) {
  const long long i = (long long)blockIdx.x * blockDim.x + threadIdx.x;
  if (i >= total) return;
  const int e = (int)(i >> 3), h = (int)(i & 7);
  const int t = tg[e], s = sc[e];
  const float v    = edge_logit(a_self, a_adjc, t, s, h);
  const float ex   = __expf(v - m[t * HH + h]);
  const float attn = ex / (m2[t * HH + h] + EPSV);

  // deterministic stateless hash -> bernoulli(0.5), inverted dropout
  unsigned x = (unsigned)i * 0x9E3779B9u + 42u;
  x ^= x >> 16; x *= 0x7feb352du;
  x ^= x >> 15; x *= 0x846ca68bu;
  x ^= x >> 16;
  out[i] = (x & 1u) ? attn * 2.0f : 0.0f;
}

extern "C" void kernel_launch(void* const* d_in, const int* in_sizes, int n_in,
                              void* d_out, int out_size, void* d_ws, size_t ws_size,
                              hipStream_t stream) {
  const float* X  = (const float*)d_in[0];
  const float* ks = (const float*)d_in[1];
  const float* ka = (const float*)d_in[2];
  const int*   tg = (const int*)d_in[4];
  const int*   sc = (const int*)d_in[5];
  const int E = in_sizes[4];   // edges (B=1)
  const int N = in_sizes[6];   // nodes (degree is [B,N])
  const int R = N * HH;

  float* a_self = (float*)d_ws;   // [R]
  float* a_adjc = a_self + R;     // [R]
  float* m      = a_adjc + R;     // [R]
  float* m2     = m + R;          // [R]  (m and m2 contiguous)
  float* out    = (float*)d_out;  // [E*H]

  // Pass 1: WMMA attention logits
  const int tiles  = R >> 4;
  int blocksW = (tiles + 7) / 8;  // 8 waves (256 threads) per block
  if (blocksW < 1) blocksW = 1;
  if (blocksW > 65535) blocksW = 65535;
  gat_logits_wmma<<<blocksW, 256, 0, stream>>>(X, ks, ka, a_self, a_adjc, R);

  // init m and m2 to -inf (segment_max identity)
  init_neg_inf<<<(2 * R + 255) / 256, 256, 0, stream>>>(m, 2 * R);

  const long long total = (long long)E * HH;
  const int blocksE = (int)((total + 255) / 256);
  edge_max1<<<blocksE, 256, 0, stream>>>(tg, sc, a_self, a_adjc, m, total);
  edge_max2<<<blocksE, 256, 0, stream>>>(tg, sc, a_self, a_adjc, m, m2, total);
  edge_out <<<blocksE, 256, 0, stream>>>(tg, sc, a_self, a_adjc, m, m2, out, total);
}